// CrossCategoryLoss_25400436588776
// MI455X (gfx1250) — compile-verified
//
#include <hip/hip_runtime.h>

// ---------------------------------------------------------------------------
// CrossCategoryLoss for MI455X (gfx1250, wave32).
// Memory-bound streaming kernel: 100 B/row traffic, ~18.7 us floor @ 23.3 TB/s.
// One row per lane; NT-hinted b128 loads; base-2 transcendentals
// (v_exp_f32 / v_log_f32 / v_rcp_f32); final ln(2) rescale exploits
// relu positive-homogeneity.
// ---------------------------------------------------------------------------

typedef __attribute__((ext_vector_type(4))) float v4f;

__device__ __forceinline__ float dev_exp2(float x) { return __builtin_amdgcn_exp2f(x); }
__device__ __forceinline__ float dev_log2(float x) { return __builtin_amdgcn_logf(x); }   // v_log_f32 = log2
__device__ __forceinline__ float dev_rcp (float x) { return __builtin_amdgcn_rcpf(x); }

// Term table: (A index into log_a, B index into log_b, G index into gamma,
// NOT flag selects log(1-p) vs log(p)).
static constexpr int T_A[36]   = {0,0,0,0,0,0, 1,1,1,1,1,1, 2,2,2,2,2,2,2,2,2,
                                  4,4,4,4,4,4, 5,5,5,5,5,5, 6,7,7};
static constexpr int T_B[36]   = {4,4,4,6,6,6, 5,5,5,6,6,6, 4,4,4,5,5,5,6,7,7,
                                  0,0,0,2,2,2, 1,1,1,2,2,2, 2,2,2};
static constexpr int T_G[36]   = {4,1,2,4,1,2, 5,0,2,5,0,2, 4,1,2,5,0,2,6,7,2,
                                  4,1,2,4,1,2, 5,0,2,5,0,2, 6,7,2};
static constexpr int T_NOT[36] = {0,1,1,0,1,1, 0,1,1,0,1,1, 0,1,1,0,1,1,0,0,1,
                                  0,1,1,0,1,1, 0,1,1,0,1,1, 0,0,1};

__global__ __launch_bounds__(256)
void cross_category_loss_kernel(const float* __restrict__ alpha,
                                const float* __restrict__ beta,
                                const float* __restrict__ gamma,
                                float* __restrict__ out,
                                int n_rows)
{
    const int r = blockIdx.x * blockDim.x + threadIdx.x;
    if (r >= n_rows) return;

    constexpr float LOG2E = 1.44269504088896340736f;
    constexpr float LN2   = 0.69314718055994530942f;
    constexpr float EPS   = 1e-8f;

    // ---- Streaming, NT-hinted vector loads (2x b128 per input) ------------
    const v4f* __restrict__ a4 = (const v4f*)alpha + 2 * (size_t)r;
    const v4f* __restrict__ b4 = (const v4f*)beta  + 2 * (size_t)r;
    const v4f* __restrict__ g4 = (const v4f*)gamma + 2 * (size_t)r;

    v4f aL = __builtin_nontemporal_load(a4);
    v4f aH = __builtin_nontemporal_load(a4 + 1);
    v4f bL = __builtin_nontemporal_load(b4);
    v4f bH = __builtin_nontemporal_load(b4 + 1);
    v4f gL = __builtin_nontemporal_load(g4);
    v4f gH = __builtin_nontemporal_load(g4 + 1);

    // Scale logits to base-2 units once.
    float ax[8], bx[8], gx[8];
#pragma unroll
    for (int i = 0; i < 4; ++i) {
        ax[i]     = aL[i] * LOG2E;  ax[i + 4] = aH[i] * LOG2E;
        bx[i]     = bL[i] * LOG2E;  bx[i + 4] = bH[i] * LOG2E;
        gx[i]     = gL[i] * LOG2E;  gx[i + 4] = gH[i] * LOG2E;
    }

    // ---- log2-softmax for alpha ------------------------------------------
    float la[8], lb[8], lg[8], lng[8];
    {
        float m = ax[0];
#pragma unroll
        for (int i = 1; i < 8; ++i) m = fmaxf(m, ax[i]);
        float s = 0.f;
#pragma unroll
        for (int i = 0; i < 8; ++i) s += dev_exp2(ax[i] - m);
        float lse = m + dev_log2(s);
#pragma unroll
        for (int i = 0; i < 8; ++i) la[i] = ax[i] - lse;
    }
    // ---- log2-softmax for beta -------------------------------------------
    {
        float m = bx[0];
#pragma unroll
        for (int i = 1; i < 8; ++i) m = fmaxf(m, bx[i]);
        float s = 0.f;
#pragma unroll
        for (int i = 0; i < 8; ++i) s += dev_exp2(bx[i] - m);
        float lse = m + dev_log2(s);
#pragma unroll
        for (int i = 0; i < 8; ++i) lb[i] = bx[i] - lse;
    }
    // ---- gamma: log2-softmax, reuse numerators for p, then log2(1-p) ------
    {
        float m = gx[0];
#pragma unroll
        for (int i = 1; i < 8; ++i) m = fmaxf(m, gx[i]);
        float e[8];
        float s = 0.f;
#pragma unroll
        for (int i = 0; i < 8; ++i) { e[i] = dev_exp2(gx[i] - m); s += e[i]; }
        float l2s  = dev_log2(s);
        float rcps = dev_rcp(s);
#pragma unroll
        for (int i = 0; i < 8; ++i) {
            lg[i] = gx[i] - m - l2s;                      // log2 p_i
            float p = e[i] * rcps;                        // p_i = exp2(lg[i])
            lng[i] = dev_log2(fmaxf(1.0f - p, EPS));      // log2(clip(1-p))
        }
    }

    // ---- 36 fully-unrolled relu terms (indices constant-fold) -------------
    float acc = 0.f;
#pragma unroll
    for (int t = 0; t < 36; ++t) {
        const float c = T_NOT[t] ? lng[T_G[t]] : lg[T_G[t]];
        acc += fmaxf(la[T_A[t]] + lb[T_B[t]] - c, 0.0f);
    }

    // Everything above is in log2 units; relu(k*x)=k*relu(x) for k>0,
    // so one final scale by ln(2) recovers natural-log units.
    __builtin_nontemporal_store(acc * LN2, out + r);
}

extern "C" void kernel_launch(void* const* d_in, const int* in_sizes, int n_in,
                              void* d_out, int out_size, void* d_ws, size_t ws_size,
                              hipStream_t stream) {
    const float* alpha = (const float*)d_in[0];
    const float* beta  = (const float*)d_in[1];
    const float* gamma = (const float*)d_in[2];
    float* out = (float*)d_out;

    const int n_rows = in_sizes[0] / 8;           // B = 4,194,304
    const int block  = 256;                       // 8 wave32 waves
    const int grid   = (n_rows + block - 1) / block;

    cross_category_loss_kernel<<<grid, block, 0, stream>>>(alpha, beta, gamma, out, n_rows);
}